// GeometricRotation_45243185496829
// MI455X (gfx1250) — compile-verified
//
#include <hip/hip_runtime.h>

// ---------------------------------------------------------------------------
// GeometricRotation on MI455X (gfx1250, wave32, WMMA)
//
//   gens -> skew(0.5*(G-G^T))/2^s -> expm via Horner Taylor (deg 8) with
//   fused (A@P)*(1/k)+I GEMMs -> square s times -> fold R1@R2@R3 -> x@Rtot.
//
// GEMM v2: K-pair interleaved B in LDS (single ds_load_b64 per B frag),
// stride-36 A tile (b128 staging stores, conflict-free b64 frag reads),
// register-double-buffered global staging to overlap VMEM with WMMA.
// ---------------------------------------------------------------------------

typedef float v2f __attribute__((ext_vector_type(2)));
typedef float v8f __attribute__((ext_vector_type(8)));

#define GDIM 1024
#define BM 128
#define BN 128
#define BK 32
#define LDS_A_STRIDE 36                 // %4==0 -> b128 stores; conflict-free col reads
#define SB_PAIR_STRIDE (2 * BN + 32)    // 288: +32 floats -> lane halves on disjoint banks

// S[r][c] = (G[r][c] - G[c][r]) * scale    (scale = 0.5 / 2^s)
__global__ __launch_bounds__(256)
void geo_skew_scale(const float* __restrict__ g, float* __restrict__ s, float scale) {
    int idx = blockIdx.x * blockDim.x + threadIdx.x;   // over GDIM*GDIM
    int r = idx >> 10;
    int c = idx & (GDIM - 1);
    s[idx] = (g[idx] - g[c * GDIM + r]) * scale;
}

// P = alpha*A + I
__global__ __launch_bounds__(256)
void geo_scale_add_identity(const float* __restrict__ a, float* __restrict__ p, float alpha) {
    int idx = blockIdx.x * blockDim.x + threadIdx.x;
    int r = idx >> 10;
    int c = idx & (GDIM - 1);
    p[idx] = a[idx] * alpha + ((r == c) ? 1.0f : 0.0f);
}

// Stage one BMxBK A tile and BKxBN B tile from global into registers.
__device__ __forceinline__
void geo_load_tile_regs(const float* __restrict__ A, const float* __restrict__ B,
                        int N, int K, int blockM, int blockN, int k0, int tid,
                        float4 aReg[4], v2f bReg[8]) {
    #pragma unroll
    for (int j = 0; j < 4; ++j) {
        int chunk = tid + j * 256;        // 0..1023, 4 floats each
        int row = chunk >> 3;             // 8 chunks per 32-wide A row
        int cg  = chunk & 7;
        aReg[j] = *(const float4*)(A + (size_t)(blockM + row) * K + k0 + cg * 4);
    }
    #pragma unroll
    for (int j = 0; j < 8; ++j) {
        int p  = tid + j * 256;           // pair id, 0..2047
        int kp = p >> 7;                  // k-pair row (0..15)
        int n  = p & 127;
        const float* src = B + (size_t)(k0 + kp * 2) * N + blockN + n;
        v2f b;
        b.x = src[0];                     // B[k][n]
        b.y = src[N];                     // B[k+1][n]
        bReg[j] = b;
    }
}

__device__ __forceinline__
void geo_store_tile_lds(float* __restrict__ sA, float* __restrict__ sB, int tid,
                        const float4 aReg[4], const v2f bReg[8]) {
    #pragma unroll
    for (int j = 0; j < 4; ++j) {
        int chunk = tid + j * 256;
        int row = chunk >> 3;
        int cg  = chunk & 7;
        *(float4*)(&sA[row * LDS_A_STRIDE + cg * 4]) = aReg[j];   // 16B aligned
    }
    #pragma unroll
    for (int j = 0; j < 8; ++j) {
        int p  = tid + j * 256;
        int kp = p >> 7;
        int n  = p & 127;
        *(v2f*)(&sB[kp * SB_PAIR_STRIDE + n * 2]) = bReg[j];      // 8B aligned, conflict-free
    }
}

// C[M,N] = alpha * (A[M,K] @ B[K,N]) (+ I if addIdentity)
// Row-major, M%128==0, N%128==0, K%32==0.
__global__ __launch_bounds__(256)
void geo_gemm_f32_wmma(const float* __restrict__ A, const float* __restrict__ B,
                       float* __restrict__ C, int M, int N, int K,
                       float alpha, int addIdentity) {
    __shared__ __align__(16) float sA[BM * LDS_A_STRIDE];             // 18432 B
    __shared__ __align__(16) float sB[(BK / 2) * SB_PAIR_STRIDE];     // 18432 B

    const int tid   = threadIdx.x;
    const int wave  = tid >> 5;
    const int lane  = tid & 31;
    const int lmod  = lane & 15;
    const int lhalf = lane >> 4;

    const int mbase = (wave >> 1) * 32;   // wave row base inside block tile
    const int nbase = (wave & 1) * 64;    // wave col base inside block tile

    const int blockM = blockIdx.y * BM;
    const int blockN = blockIdx.x * BN;

    v8f zero = {};
    v8f acc[2][4];
    #pragma unroll
    for (int mt = 0; mt < 2; ++mt)
        #pragma unroll
        for (int nt = 0; nt < 4; ++nt)
            acc[mt][nt] = zero;

    float4 aReg[4];
    v2f    bReg[8];

    // prologue: stage tile 0
    geo_load_tile_regs(A, B, N, K, blockM, blockN, 0, tid, aReg, bReg);
    geo_store_tile_lds(sA, sB, tid, aReg, bReg);
    __syncthreads();

    for (int k0 = 0; k0 < K; k0 += BK) {
        const bool hasNext = (k0 + BK) < K;
        // issue next tile's global loads while WMMAs run on the current tile
        if (hasNext)
            geo_load_tile_regs(A, B, N, K, blockM, blockN, k0 + BK, tid, aReg, bReg);

        #pragma unroll
        for (int kk = 0; kk < BK; kk += 4) {
            // A frag: lane holds A[m][kf], A[m][kf+1]; m = mbase+mt*16+lmod,
            // kf = kk + 2*lhalf   (ISA 7.12.2 fp32 16x4 A layout)
            v2f afrag[2];
            #pragma unroll
            for (int mt = 0; mt < 2; ++mt) {
                int m = mbase + mt * 16 + lmod;
                afrag[mt] = *(const v2f*)(&sA[m * LDS_A_STRIDE + kk + 2 * lhalf]);
            }
            // B frag: pair-interleaved -> one b64 per tile:
            // {B[kf][n], B[kf+1][n]} contiguous at sB[(kf/2)*288 + 2n]
            const int pr = (kk + 2 * lhalf) >> 1;
            v2f bfrag[4];
            #pragma unroll
            for (int nt = 0; nt < 4; ++nt) {
                int n = nbase + nt * 16 + lmod;
                bfrag[nt] = *(const v2f*)(&sB[pr * SB_PAIR_STRIDE + n * 2]);
            }
            #pragma unroll
            for (int mt = 0; mt < 2; ++mt)
                #pragma unroll
                for (int nt = 0; nt < 4; ++nt)
                    acc[mt][nt] = __builtin_amdgcn_wmma_f32_16x16x4_f32(
                        false, afrag[mt], false, bfrag[nt],
                        (short)0, acc[mt][nt], false, false);
        }
        __syncthreads();
        if (hasNext) {
            geo_store_tile_lds(sA, sB, tid, aReg, bReg);
            __syncthreads();
        }
    }

    // epilogue: C/D layout — VGPR r holds rows r (lanes 0-15) / r+8 (lanes 16-31)
    #pragma unroll
    for (int mt = 0; mt < 2; ++mt) {
        #pragma unroll
        for (int nt = 0; nt < 4; ++nt) {
            int n = blockN + nbase + nt * 16 + lmod;
            #pragma unroll
            for (int r = 0; r < 8; ++r) {
                int m = blockM + mbase + mt * 16 + r + 8 * lhalf;
                float val = acc[mt][nt][r] * alpha;
                if (addIdentity && (m == n)) val += 1.0f;
                C[(size_t)m * N + n] = val;
            }
        }
    }
}

extern "C" void kernel_launch(void* const* d_in, const int* in_sizes, int n_in,
                              void* d_out, int out_size, void* d_ws, size_t ws_size,
                              hipStream_t stream) {
    const float* x    = (const float*)d_in[0];   // [4,4096,1024] -> 16384 x 1024
    const float* gens = (const float*)d_in[1];   // [3,1024,1024]
    float* out = (float*)d_out;                  // 16384 x 1024
    float* ws  = (float*)d_ws;

    const size_t DD = (size_t)GDIM * GDIM;
    float* bufA  = ws;            // scaled skew generator
    float* bufP  = ws + DD;       // Horner / squaring ping
    float* bufT  = ws + 2 * DD;   // Horner / squaring pong
    float* bufR  = ws + 3 * DD;   // running rotation product
    float* bufR2 = ws + 4 * DD;   // alternate product buffer

    const int XROWS  = 4 * 4096;          // 16384
    const int SQ     = 5;                 // scaling-and-squaring steps
    const float SCL  = 0.5f / 32.0f;      // 0.5 (skew) * 2^-SQ
    const int MTERMS = 8;                 // Taylor degree

    dim3 blk(256);
    dim3 gridE((unsigned)(DD / 256));
    dim3 gridG(GDIM / BN, GDIM / BM);     // 8 x 8

    for (int i = 0; i < 3; ++i) {
        geo_skew_scale<<<gridE, blk, 0, stream>>>(gens + (size_t)i * DD, bufA, SCL);
        geo_scale_add_identity<<<gridE, blk, 0, stream>>>(bufA, bufP, 1.0f / (float)MTERMS);

        float* P = bufP;
        float* T = bufT;
        // Horner: P <- I + (A/k) @ P,  k = MTERMS-1 .. 1
        for (int k = MTERMS - 1; k >= 1; --k) {
            geo_gemm_f32_wmma<<<gridG, blk, 0, stream>>>(bufA, P, T, GDIM, GDIM, GDIM,
                                                         1.0f / (float)k, 1);
            float* tmp = P; P = T; T = tmp;
        }
        // squaring: P <- P @ P, SQ times
        for (int j = 0; j < SQ; ++j) {
            geo_gemm_f32_wmma<<<gridG, blk, 0, stream>>>(P, P, T, GDIM, GDIM, GDIM,
                                                         1.0f, 0);
            float* tmp = P; P = T; T = tmp;
        }
        // P holds R_i; fold into running product
        if (i == 0) {
            hipMemcpyAsync(bufR, P, DD * sizeof(float), hipMemcpyDeviceToDevice, stream);
        } else if (i == 1) {
            geo_gemm_f32_wmma<<<gridG, blk, 0, stream>>>(bufR, P, bufR2, GDIM, GDIM, GDIM,
                                                         1.0f, 0);
        } else {
            geo_gemm_f32_wmma<<<gridG, blk, 0, stream>>>(bufR2, P, bufR, GDIM, GDIM, GDIM,
                                                         1.0f, 0);
        }
    }

    // out = x @ Rtot   (single 16384x1024x1024 GEMM)
    dim3 gridX(GDIM / BN, XROWS / BM);    // 8 x 128
    geo_gemm_f32_wmma<<<gridX, blk, 0, stream>>>(x, bufR, out, XROWS, GDIM, GDIM,
                                                 1.0f, 0);
}